// LocalCostVolume_66606352827284
// MI455X (gfx1250) — compile-verified
//
#include <hip/hip_runtime.h>

// LocalCostVolume disparity refinement for MI455X (gfx1250, wave32).
//
// Gather-bandwidth-bound (~0.5 GB effective scattered traffic vs ~0.4 GFLOP):
//   - one thread per pixel
//   - 10-wide disparity window loaded once with non-temporal global loads
//     (covers all 22 floor/ceil gathers of the 11 softmax samples -> 2.2x
//     less HBM traffic than per-sample gathering)
//   - window parked in a private LDS slice; the 22 data-dependent lookups
//     become ds_load_b32 (same-wave LDS ordering => no barrier), replacing
//     ~400 v_cndmask select ops per thread with ~25 ds ops
//   - native v_exp_f32 softmax, fused weighted-sum

namespace {

constexpr int Dc = 192, Hc = 320, Wc = 640;
constexpr int HW = Hc * Wc;
constexpr int SAMPLES = 11;    // SAMPLE_POINTS + 1
constexpr int WIN = 10;        // max distinct integer d-indices in the +/-4 window
constexpr int SSTRIDE = 12;    // LDS dwords per thread: 16B-aligned slices,
                               // 12*L mod 64 -> at worst 2-way bank conflicts
constexpr int BLOCK = 256;     // 8 wave32 waves

} // namespace

__global__ __launch_bounds__(BLOCK) void
local_cost_volume_kernel(const float* __restrict__ vol,   // [B, D, H, W]
                         const float* __restrict__ disp,  // [B, 1, H, W]
                         float* __restrict__ out,         // [B, 1, H, W]
                         int npix) {
  __shared__ float smem[BLOCK * SSTRIDE];

  const int idx = blockIdx.x * blockDim.x + threadIdx.x;
  if (idx >= npix) return;

  const int b  = idx / HW;
  const int hw = idx - b * HW;
  // Column through the disparity axis for this (b, h, w); d-stride = HW floats.
  const float* __restrict__ col = vol + (size_t)b * Dc * HW + hw;

  const float d        = disp[idx];
  const float lower    = fmaxf(d - 4.0f, 0.0f);
  const float upper    = fminf(fmaxf(d + 4.0f, 0.0f), (float)Dc);
  const float interval = (upper - lower) * 0.1f;

  // lower >= 0 so trunc == floor; lower <= d < 191 -> d0 in [0, 190].
  const int d0 = (int)lower;

  // ---- Load the 10-slice window once (10 independent NT loads, stride HW).
  // All indices needed by all 11 samples provably land in [d0, d0+9].
  float w[WIN];
#pragma unroll
  for (int j = 0; j < WIN; ++j) {
    int dj = d0 + j;
    dj = dj > (Dc - 1) ? (Dc - 1) : dj;          // clamp: dup-loads hit cache
    w[j] = __builtin_nontemporal_load(col + (size_t)dj * HW);
  }

  // ---- Park the window in this thread's private LDS slice (wide stores).
  const int sbase = threadIdx.x * SSTRIDE;       // byte offset sbase*4, 16B-aligned
  float* __restrict__ slice = &smem[sbase];
  {
    float4 a = make_float4(w[0], w[1], w[2], w[3]);
    float4 c = make_float4(w[4], w[5], w[6], w[7]);
    float2 e = make_float2(w[8], w[9]);
    *reinterpret_cast<float4*>(slice + 0) = a;   // ds_store_b128
    *reinterpret_cast<float4*>(slice + 4) = c;   // ds_store_b128
    *reinterpret_cast<float2*>(slice + 8) = e;   // ds_store_b64
  }
  // Same-wave LDS ops are in-order and the slice is thread-private: no barrier.

  // ---- 11 samples: ds_load_b32 gathers + linear interpolation.
  // val = v[fi] + (v[min(fi+1,191)] - v[fi]) * (cand - fi)
  // == reference's vf*floor_rate + vc*ceil_rate for fractional, integer,
  // and top-clamped (cand in (191,192], both entries = v[191]) cases.
  float vals[SAMPLES], cands[SAMPLES];
  float m = -3.402823466e38f;
#pragma unroll
  for (int s = 0; s < SAMPLES; ++s) {
    const float cand = lower + (float)s * interval;
    int fi = (int)cand;                          // cand >= 0: trunc == floor
    fi = fi > (Dc - 1) ? (Dc - 1) : fi;
    const float frac = cand - (float)fi;
    const int o  = fi - d0;                                      // in [0, 8]
    const int o2 = ((fi + 1) > (Dc - 1) ? (Dc - 1) : (fi + 1)) - d0; // in [0, 9]
    const float vf = slice[o];                   // ds_load_b32
    const float vc = slice[o2];                  // ds_load_b32
    const float val = vf + (vc - vf) * frac;
    vals[s]  = val;
    cands[s] = cand;
    m = fmaxf(m, val);
  }

  // ---- Softmax-weighted expected disparity (v_exp_f32 per sample).
  float esum = 0.0f, acc = 0.0f;
#pragma unroll
  for (int s = 0; s < SAMPLES; ++s) {
    const float e = __expf(vals[s] - m);
    esum += e;
    acc  = fmaf(e, cands[s], acc);
  }
  out[idx] = acc / esum;
}

extern "C" void kernel_launch(void* const* d_in, const int* in_sizes, int n_in,
                              void* d_out, int out_size, void* d_ws, size_t ws_size,
                              hipStream_t stream) {
  (void)in_sizes; (void)n_in; (void)d_ws; (void)ws_size;
  const float* vol  = (const float*)d_in[0];  // old_cost_volume [4,192,320,640] f32
  const float* disp = (const float*)d_in[1];  // cur_disparity   [4,1,320,640]  f32
  float* out        = (float*)d_out;          // [4,1,320,640] f32

  const int npix   = out_size;                // B*H*W = 819200 (= 3200 * 256)
  const int blocks = (npix + BLOCK - 1) / BLOCK;
  hipLaunchKernelGGL(local_cost_volume_kernel, dim3(blocks), dim3(BLOCK), 0,
                     stream, vol, disp, out, npix);
}